// PostModel_6425271074838
// MI455X (gfx1250) — compile-verified
//
#include <hip/hip_runtime.h>
#include <hip/hip_bf16.h>
#include <stdint.h>

#define N_ANCHORS   33600
#define N_PAD       65536
#define NUM_CLASSES 80
#define MAX_DET     4096
#define CONF_THRE   0.3f
#define NMS_THRE    0.65f
#define CLS_OFFSET  4.0f
#define SUP_WORDS   128     // 4096 / 32

typedef __attribute__((ext_vector_type(2))) float v2f;
typedef __attribute__((ext_vector_type(8))) float v8f;
typedef __attribute__((ext_vector_type(4))) unsigned int v4u;
typedef __attribute__((ext_vector_type(8))) unsigned int v8u;

// ---------------------------------------------------------------------------
// Kernel 1: decode boxes, score = obj * max(cls), build 64-bit sort keys.
// key = (ordered(masked_score) << 32) | (0xFFFFFFFF - idx)  -> descending sort
// reproduces lax.top_k (value desc, index asc on ties). Pad keys = 0.
// ---------------------------------------------------------------------------
__global__ void decode_kernel(const float* __restrict__ raw,
                              unsigned long long* __restrict__ keys,
                              float4* __restrict__ boxes,
                              int* __restrict__ cls) {
    int i = blockIdx.x * blockDim.x + threadIdx.x;
    if (i >= N_PAD) return;
    if (i >= N_ANCHORS) { keys[i] = 0ULL; return; }

    const float* p = raw + (size_t)i * (5 + NUM_CLASSES);
    float cx = p[0], cy = p[1], w = p[2], h = p[3], obj = p[4];

    float best = p[5];
    int bc = 0;
    #pragma unroll 4
    for (int c = 1; c < NUM_CLASSES; ++c) {
        float v = p[5 + c];
        if (v > best) { best = v; bc = c; }   // first-max == jnp.argmax
    }
    float score  = obj * best;
    float masked = (score >= CONF_THRE) ? score : -1.0f;

    boxes[i] = make_float4(cx - 0.5f * w, cy - 0.5f * h,
                           cx + 0.5f * w, cy + 0.5f * h);
    cls[i] = bc;

    unsigned int u = __float_as_uint(masked);
    u = (u & 0x80000000u) ? ~u : (u | 0x80000000u);   // order-preserving map
    keys[i] = ((unsigned long long)u << 32) |
              (unsigned long long)(0xFFFFFFFFu - (unsigned int)i);
}

// ---------------------------------------------------------------------------
// Kernel 2a: global bitonic compare-exchange pass (descending), j >= 2048.
// ---------------------------------------------------------------------------
__global__ void bitonic_pass(unsigned long long* __restrict__ keys, int j, int k) {
    int i = blockIdx.x * blockDim.x + threadIdx.x;
    int l = i ^ j;
    if (l > i) {
        unsigned long long a = keys[i], b = keys[l];
        bool up = (i & k) == 0;
        bool sw = up ? (a < b) : (a > b);      // descending overall
        if (sw) { keys[i] = b; keys[l] = a; }
    }
}

// ---------------------------------------------------------------------------
// Kernel 2b: in-LDS bitonic sort of a 2048-key block (stages k = 2..2048).
// 1024 threads, each does one compare-exchange per pass; 16 KB LDS.
// ---------------------------------------------------------------------------
__global__ void bitonic_local_sort(unsigned long long* __restrict__ keys) {
    __shared__ unsigned long long sk[2048];
    const int base = blockIdx.x * 2048;
    for (int t = threadIdx.x; t < 2048; t += 1024) sk[t] = keys[base + t];
    __syncthreads();
    for (int k = 2; k <= 2048; k <<= 1) {
        for (int j = k >> 1; j > 0; j >>= 1) {
            int t = threadIdx.x;
            int i = ((t & ~(j - 1)) << 1) | (t & (j - 1));
            int l = i | j;
            unsigned long long a = sk[i], b = sk[l];
            bool up = (((base + i) & k) == 0);
            bool sw = up ? (a < b) : (a > b);
            if (sw) { sk[i] = b; sk[l] = a; }
            __syncthreads();
        }
    }
    for (int t = threadIdx.x; t < 2048; t += 1024) keys[base + t] = sk[t];
}

// ---------------------------------------------------------------------------
// Kernel 2c: in-LDS tail of a merge stage (j = 1024..1) for k >= 4096.
// Direction is uniform per 2048-block since k >= 2*blockSize.
// ---------------------------------------------------------------------------
__global__ void bitonic_local_merge(unsigned long long* __restrict__ keys, int k) {
    __shared__ unsigned long long sk[2048];
    const int base = blockIdx.x * 2048;
    for (int t = threadIdx.x; t < 2048; t += 1024) sk[t] = keys[base + t];
    __syncthreads();
    const bool up = ((base & k) == 0);
    for (int j = 1024; j > 0; j >>= 1) {
        int t = threadIdx.x;
        int i = ((t & ~(j - 1)) << 1) | (t & (j - 1));
        int l = i | j;
        unsigned long long a = sk[i], b = sk[l];
        bool sw = up ? (a < b) : (a > b);
        if (sw) { sk[i] = b; sk[l] = a; }
        __syncthreads();
    }
    for (int t = threadIdx.x; t < 2048; t += 1024) keys[base + t] = sk[t];
}

// ---------------------------------------------------------------------------
// Kernel 3: gather top-4096 candidates, build class-offset boxes (SoA,
// contiguous arrays at 4096-element stride for the TDM descriptor) + areas,
// candidate output rows [x1,y1,x2,y2,conf], and valid bitmask.
// ---------------------------------------------------------------------------
__global__ void gather_kernel(const unsigned long long* __restrict__ keys,
                              const float4* __restrict__ boxes,
                              const int* __restrict__ cls,
                              float* __restrict__ SOA,   // 5 arrays of 4096 floats
                              float* __restrict__ cand,
                              unsigned int* __restrict__ validw) {
    int t = blockIdx.x * blockDim.x + threadIdx.x;
    if (t >= MAX_DET) return;

    unsigned long long key = keys[t];
    unsigned int lo = (unsigned int)(key & 0xFFFFFFFFu);
    unsigned int hi = (unsigned int)(key >> 32);
    int idx = (int)(0xFFFFFFFFu - lo);
    float conf = (hi & 0x80000000u) ? __uint_as_float(hi ^ 0x80000000u)
                                    : __uint_as_float(~hi);

    float4 b = boxes[idx];
    float off = (float)cls[idx] * CLS_OFFSET;
    SOA[0 * MAX_DET + t] = b.x + off;                 // X1
    SOA[1 * MAX_DET + t] = b.y + off;                 // Y1
    SOA[2 * MAX_DET + t] = b.z + off;                 // X2
    SOA[3 * MAX_DET + t] = b.w + off;                 // Y2
    SOA[4 * MAX_DET + t] = (b.z - b.x) * (b.w - b.y); // AREA (offset cancels)

    float* c5 = cand + (size_t)t * 5;
    c5[0] = b.x; c5[1] = b.y; c5[2] = b.z; c5[3] = b.w; c5[4] = conf;

    bool valid = conf >= 0.0f;
    unsigned int bal = __builtin_amdgcn_ballot_w32(valid);
    if ((threadIdx.x & 31u) == 0u) validw[t >> 5] = bal;
}

// ---------------------------------------------------------------------------
// Kernel 4: suppression bitmask. One wave32 per (16-row x 32-col) tile.
//  - Row strip (16 elems x 5 arrays, stride 4096 floats) is DMA'd into LDS
//    with one TDM tensor_load_to_lds (2-D descriptor), synced by TENSORcnt.
//  - union_base[i][j] = area_i + area_j via V_WMMA_F32_16X16X4_F32 as the
//    rank-2 product [area_i,1] x [[1],[area_j]].
//  - inter/compare in VALU; bits packed with wave32 ballots -> one full
//    32-bit suppression word per row, plain store (no atomics, no zeroing).
// ---------------------------------------------------------------------------
__global__ void iou_kernel(const float* __restrict__ SOA,
                           unsigned int* __restrict__ sup) {
    const int lane = threadIdx.x;                 // 0..31
    const int bx = blockIdx.x & (SUP_WORDS - 1);  // word column (32 cols)
    const int by = blockIdx.x >> 7;               // 16-row tile
    const int ri0 = by * 16;
    const int cj0 = bx * 32;
    const int m   = lane & 15;

    // Only shared allocation in this kernel -> LDS offset 0.
    __shared__ float stile[5 * 16];   // rows: X1, Y1, X2, Y2, AREA

    // ---- TDM: 2-D tile (tile_dim0=16 elems, tile_dim1=5 rows), dword data,
    //      tensor_dim0_stride = 4096 elements, dest LDS addr 0. ----
    {
        unsigned long long ga =
            (unsigned long long)(uintptr_t)(SOA + ri0);      // tile start
        const unsigned int TD0 = 1u << 20;                   // ample dims: no OOB clip
        const unsigned int TD1 = 64u;
        v4u g0;
        g0.x = 1u;                                           // count=1, user mode
        g0.y = 0u;                                           // lds_addr = 0
        g0.z = (unsigned int)ga;                             // global_addr[31:0]
        g0.w = ((unsigned int)(ga >> 32) & 0x01FFFFFFu) | 0x80000000u; // type=2
        v8u g1;
        g1.s0 = 0x00020000u;                                 // data_size=4B, mask=0
        g1.s1 = (TD0 & 0xFFFFu) << 16;                       // tensor_dim0 lo16
        g1.s2 = ((TD0 >> 16) & 0xFFFFu) | ((TD1 & 0xFFFFu) << 16);
        g1.s3 = ((TD1 >> 16) & 0xFFFFu) | (16u << 16);       // tile_dim0 = 16
        g1.s4 = 5u;                                          // tile_dim1=5, dim2=0
        g1.s5 = 4096u;                                       // dim0_stride lo32
        g1.s6 = 0u;                                          // dim0_stride hi16
        g1.s7 = 0u;
        asm volatile("tensor_load_to_lds %0, %1"
                     :
                     : "s"(g0), "s"(g1)
                     : "memory");
        __builtin_amdgcn_s_wait_tensorcnt(0);
    }
    __syncthreads();

    const float* sx1 = stile +  0;
    const float* sy1 = stile + 16;
    const float* sx2 = stile + 32;
    const float* sy2 = stile + 48;
    const float* sar = stile + 64;

    const float* X1   = SOA + 0 * MAX_DET;
    const float* Y1   = SOA + 1 * MAX_DET;
    const float* X2   = SOA + 2 * MAX_DET;
    const float* Y2   = SOA + 3 * MAX_DET;
    const float* AREA = SOA + 4 * MAX_DET;

    // A (16x4 f32): lanes 0-15 hold K=0,1 ; lanes 16-31 hold K=2,3 (zeros)
    v2f a;
    if (lane < 16) { a.x = sar[m]; a.y = 1.0f; }
    else           { a.x = 0.0f;   a.y = 0.0f; }

    unsigned int hb[2][8];

    #pragma unroll
    for (int h = 0; h < 2; ++h) {
        const int col = cj0 + h * 16 + m;
        const float jx1 = X1[col], jy1 = Y1[col];
        const float jx2 = X2[col], jy2 = Y2[col];
        const float jar = AREA[col];

        // B (4x16 f32): lanes 0-15 hold K=0,1 rows; lanes 16-31 K=2,3 (zeros)
        v2f b;
        if (lane < 16) { b.x = 1.0f; b.y = jar; }
        else           { b.x = 0.0f; b.y = 0.0f; }

        v8f c = {};
        // S[r] = area_i[row] + area_j[col];  row = r + 8*(lane>=16)
        v8f S = __builtin_amdgcn_wmma_f32_16x16x4_f32(
            false, a, false, b, (short)0, c, false, false);

        const int rbase = (lane < 16) ? 0 : 8;
        #pragma unroll
        for (int r = 0; r < 8; ++r) {
            const int row = rbase + r;
            float ix1 = fmaxf(sx1[row], jx1);
            float iy1 = fmaxf(sy1[row], jy1);
            float ix2 = fminf(sx2[row], jx2);
            float iy2 = fminf(sy2[row], jy2);
            float inter = fmaxf(ix2 - ix1, 0.0f) * fmaxf(iy2 - iy1, 0.0f);
            float uni = S[r] - inter;
            bool bit = (uni > 0.0f) && (inter > NMS_THRE * uni);
            hb[h][r] = __builtin_amdgcn_ballot_w32(bit);
        }
    }

    // Each of lanes 0-15 assembles/stores the 32-bit word for its row.
    if (lane < 16) {
        unsigned int w;
        if (lane < 8) {
            w = (hb[0][lane] & 0xFFFFu) | ((hb[1][lane] & 0xFFFFu) << 16);
        } else {
            int r = lane - 8;
            w = ((hb[0][r] >> 16) & 0xFFFFu) | (hb[1][r] & 0xFFFF0000u);
        }
        sup[(size_t)(ri0 + lane) * SUP_WORDS + bx] = w;
    }
}

// ---------------------------------------------------------------------------
// Kernel 5: sequential greedy NMS in a single wave32. Lane l owns keep-words
// [4l..4l+3]; keep-bit broadcast via ds_bpermute; next row prefetched.
// ---------------------------------------------------------------------------
__global__ void nms_kernel(const unsigned int* __restrict__ sup,
                           const unsigned int* __restrict__ validw,
                           unsigned int* __restrict__ keepw) {
    const int lane = threadIdx.x;
    unsigned int keep[4];
    #pragma unroll
    for (int c = 0; c < 4; ++c) keep[c] = validw[lane * 4 + c];

    for (int i = 0; i < MAX_DET; ++i) {
        const int w = i >> 5, owner = w >> 2, slot = w & 3;
        unsigned int kword = (unsigned int)__builtin_amdgcn_ds_bpermute(
            owner << 2, (int)keep[slot]);

        int nxt = (i + 1 < MAX_DET) ? (i + 1) : i;
        __builtin_prefetch((const void*)(sup + (size_t)nxt * SUP_WORDS + lane * 4), 0, 0);

        if ((kword >> (i & 31)) & 1u) {
            const uint4 row = *(const uint4*)(sup + (size_t)i * SUP_WORDS + lane * 4);
            const unsigned int rv[4] = { row.x, row.y, row.z, row.w };
            const int bp = i & 31;
            const unsigned int eqmask =
                (bp == 31) ? 0u : (0xFFFFFFFFu << (bp + 1));   // only j > i
            #pragma unroll
            for (int c = 0; c < 4; ++c) {
                int wj = lane * 4 + c;
                unsigned int msk = (wj < w) ? 0u
                                 : (wj > w) ? 0xFFFFFFFFu
                                            : eqmask;
                keep[c] &= ~(rv[c] & msk);
            }
        }
    }
    #pragma unroll
    for (int c = 0; c < 4; ++c) keepw[lane * 4 + c] = keep[c];
}

// ---------------------------------------------------------------------------
// Kernel 6: write output rows, zeroed where not kept.
// ---------------------------------------------------------------------------
__global__ void finalize_kernel(const float* __restrict__ cand,
                                const unsigned int* __restrict__ keepw,
                                float* __restrict__ out) {
    int t = blockIdx.x * blockDim.x + threadIdx.x;
    if (t >= MAX_DET) return;
    bool kp = (keepw[t >> 5] >> (t & 31)) & 1u;
    const float* c5 = cand + (size_t)t * 5;
    float* o5 = out + (size_t)t * 5;
    #pragma unroll
    for (int q = 0; q < 5; ++q) o5[q] = kp ? c5[q] : 0.0f;
}

// ---------------------------------------------------------------------------
extern "C" void kernel_launch(void* const* d_in, const int* in_sizes, int n_in,
                              void* d_out, int out_size, void* d_ws, size_t ws_size,
                              hipStream_t stream) {
    (void)in_sizes; (void)n_in; (void)out_size; (void)ws_size;
    const float* raw = (const float*)d_in[0];
    float* out = (float*)d_out;

    char* ws = (char*)d_ws;
    size_t off = 0;
    auto alloc = [&](size_t bytes) -> char* {
        char* p = ws + off;
        off = (off + bytes + 255) & ~(size_t)255;
        return p;
    };

    unsigned long long* keys = (unsigned long long*)alloc(N_PAD * 8);
    float4*        boxes = (float4*)alloc((size_t)N_ANCHORS * 16);
    int*           cls   = (int*)alloc((size_t)N_ANCHORS * 4);
    float*         SOA   = (float*)alloc((size_t)5 * MAX_DET * 4); // X1|Y1|X2|Y2|AREA
    float*         cand  = (float*)alloc((size_t)MAX_DET * 5 * 4);
    unsigned int*  validw = (unsigned int*)alloc(SUP_WORDS * 4);
    unsigned int*  keepw  = (unsigned int*)alloc(SUP_WORDS * 4);
    unsigned int*  sup    = (unsigned int*)alloc((size_t)MAX_DET * SUP_WORDS * 4);

    // 1) decode + keys
    decode_kernel<<<N_PAD / 256, 256, 0, stream>>>(raw, keys, boxes, cls);

    // 2) bitonic sort (descending): LDS block sort + hybrid merges (21 launches)
    bitonic_local_sort<<<N_PAD / 2048, 1024, 0, stream>>>(keys);
    for (int k = 4096; k <= N_PAD; k <<= 1) {
        for (int j = k >> 1; j >= 2048; j >>= 1)
            bitonic_pass<<<N_PAD / 256, 256, 0, stream>>>(keys, j, k);
        bitonic_local_merge<<<N_PAD / 2048, 1024, 0, stream>>>(keys, k);
    }

    // 3) gather top-4096
    gather_kernel<<<MAX_DET / 256, 256, 0, stream>>>(keys, boxes, cls,
                                                     SOA, cand, validw);

    // 4) suppression bitmask (TDM row-tile DMA + WMMA union-sum + ballots)
    iou_kernel<<<(MAX_DET / 16) * SUP_WORDS, 32, 0, stream>>>(SOA, sup);

    // 5) sequential greedy NMS (single wave32)
    nms_kernel<<<1, 32, 0, stream>>>(sup, validw, keepw);

    // 6) masked output
    finalize_kernel<<<MAX_DET / 256, 256, 0, stream>>>(cand, keepw, out);
}